// HashEncoder_64750926954715
// MI455X (gfx1250) — compile-verified
//
#include <hip/hip_runtime.h>
#include <stdint.h>

#ifndef __has_builtin
#define __has_builtin(x) 0
#endif

#define NLEV 16
#define NPOINTS 524288
#define BLOCK 256

// ---------------------------------------------------------------------------
// Compile-time per-level metadata (mirrors the Python _meta()).
// b = 1.3195079565048218, base = 16, MAX_PARAMS = 2^19.
// Levels 0..5 dense (res = 16,22,28,37,49,65), levels 6..15 spatial hash with
// table size 2^19. Values fold to immediates via template<int l> + constexpr:
// dense "% size" -> magic-multiply, fast "% 2^19" -> AND. No runtime division.
// ---------------------------------------------------------------------------
struct Meta {
  float    scale[NLEV];
  uint32_t res[NLEV];
  uint32_t res2[NLEV];   // (res^2) wrapped to u32, matches reference strides
  uint32_t size[NLEV];
  uint32_t off[NLEV];    // offset in float2 (F=2) units
  int      begin_fast;
};

__host__ __device__ constexpr Meta makeMeta() {
  Meta m = {};
  double p = 1.0;
  unsigned long long off = 0;
  int bf = NLEV;
  for (int i = 0; i < NLEV; ++i) {
    double scale = 16.0 * p - 1.0;
    long long r = (long long)scale;       // ceil for non-negative scale
    if ((double)r < scale) ++r;
    ++r;                                  // res = ceil(scale) + 1
    unsigned long long full    = (unsigned long long)r * (unsigned long long)r * (unsigned long long)r;
    unsigned long long aligned = ((full + 7ull) / 8ull) * 8ull;
    unsigned long long sz      = aligned < 524288ull ? aligned : 524288ull;
    m.scale[i] = (float)scale;
    m.res[i]   = (uint32_t)r;
    m.res2[i]  = (uint32_t)((unsigned long long)r * (unsigned long long)r);
    m.size[i]  = (uint32_t)sz;
    m.off[i]   = (uint32_t)off;
    if (full > sz && bf == NLEV) bf = i;
    off += sz;
    p *= 1.3195079565048218;
  }
  m.begin_fast = bf;
  return m;
}

constexpr Meta META = makeMeta();

// Pointer types matching the async-to-LDS builtin signature:
// arg0: addrspace(1) pointer to int __vector_size__(16); arg1: addrspace(3).
typedef int v4i_t __attribute__((vector_size(16)));
typedef __attribute__((address_space(1))) v4i_t* gas_v4i_ptr;
typedef __attribute__((address_space(3))) v4i_t* las_v4i_ptr;

// ---------------------------------------------------------------------------
// One hash level: 8 corner gathers (float2 each) + trilinear blend.
// For l==0 the caller passes an LDS-backed pointer (addrspace inferred after
// inlining -> ds_load_b64); for l>0 this is global -> global_load_b64.
// ---------------------------------------------------------------------------
template <int l>
__device__ __forceinline__ void level_accum(float px, float py, float pz,
                                            const float2* __restrict__ t2,
                                            float& a0, float& a1) {
  constexpr float    s    = META.scale[l];
  constexpr uint32_t res  = META.res[l];
  constexpr uint32_t res2 = META.res2[l];
  constexpr uint32_t sz   = META.size[l];
  constexpr uint32_t off  = META.off[l];
  constexpr bool     fast = (l >= META.begin_fast);

  const float sx  = px * s + 0.5f;
  const float sy  = py * s + 0.5f;
  const float sz_ = pz * s + 0.5f;
  const float fx = floorf(sx), fy = floorf(sy), fz = floorf(sz_);
  const float tx = sx - fx, ty = sy - fy, tz = sz_ - fz;
  const uint32_t gx = (uint32_t)fx, gy = (uint32_t)fy, gz = (uint32_t)fz;

  float acc0 = 0.0f, acc1 = 0.0f;
#pragma unroll
  for (int c = 0; c < 8; ++c) {
    const uint32_t cx = gx + (c & 1);
    const uint32_t cy = gy + ((c >> 1) & 1);
    const uint32_t cz = gz + ((c >> 2) & 1);
    uint32_t h;
    if (fast) {
      h = (cx ^ (cy * 2654435761u) ^ (cz * 805459861u)) % sz;  // AND (2^19)
    } else {
      h = (cx + cy * res + cz * res2) % sz;                    // magic-mul
    }
    const float w = ((c & 1)        ? tx : 1.0f - tx)
                  * (((c >> 1) & 1) ? ty : 1.0f - ty)
                  * (((c >> 2) & 1) ? tz : 1.0f - tz);
    const float2 f = t2[off + h];   // b64 gather (LDS for l==0)
    acc0 = fmaf(w, f.x, acc0);
    acc1 = fmaf(w, f.y, acc1);
  }
  a0 = acc0;
  a1 = acc1;
}

// ---------------------------------------------------------------------------
// Kernel: 1 point / thread, 256 threads / block (8 wave32), 2048 blocks.
// LDS (33 KB) used twice: level-0 table copy (async-loaded), then (after a
// barrier) the output transpose stage for coalesced float4 stores.
// ---------------------------------------------------------------------------
__global__ void __launch_bounds__(BLOCK)
hashenc_kernel(const float* __restrict__ pos,
               const float* __restrict__ table,
               float* __restrict__ out) {
  __shared__ float smem[BLOCK * 33];   // 8448 floats = 33 KB

  const int tid = threadIdx.x;
  const int b   = blockIdx.x * BLOCK + tid;

  // ---- Stage level-0 table (4096 x float2 = 32 KB) into LDS -------------
#if defined(__gfx1250__) && __has_builtin(__builtin_amdgcn_global_load_async_to_lds_b128)
#pragma unroll
  for (int i = 0; i < 8; ++i) {
    const int e = tid + i * BLOCK;     // float4 index, e < 2048
    __builtin_amdgcn_global_load_async_to_lds_b128(
        (gas_v4i_ptr)(table + 4 * e),
        (las_v4i_ptr)(smem + 4 * e),
        0, 0);
  }
#if __has_builtin(__builtin_amdgcn_s_wait_asynccnt)
  __builtin_amdgcn_s_wait_asynccnt(0);
#else
  asm volatile("s_wait_asynccnt 0" ::: "memory");
#endif
#else
#pragma unroll
  for (int i = 0; i < 8; ++i) {
    const int e = tid + i * BLOCK;
    reinterpret_cast<float4*>(smem)[e] = reinterpret_cast<const float4*>(table)[e];
  }
#endif
  __syncthreads();

  // ---- Load this thread's position (coalesced b32 x3) -------------------
  const float px = pos[3 * b + 0];
  const float py = pos[3 * b + 1];
  const float pz = pos[3 * b + 2];

  // ---- Level 0 from LDS --------------------------------------------------
  float l0a, l0b;
  level_accum<0>(px, py, pz, reinterpret_cast<const float2*>(smem), l0a, l0b);
  __syncthreads();                     // everyone done reading the LDS table

  float* stage = smem;                 // reuse LDS as [256][33] output stage
  stage[tid * 33 + 0] = l0a;
  stage[tid * 33 + 1] = l0b;

  // ---- Levels 1..15 from global (L2-resident table) ---------------------
  const float2* t2 = reinterpret_cast<const float2*>(table);
  float a0, a1;
#define DO_LEVEL(L)                                        \
  level_accum<L>(px, py, pz, t2, a0, a1);                  \
  stage[tid * 33 + 2 * (L) + 0] = a0;                      \
  stage[tid * 33 + 2 * (L) + 1] = a1;
  DO_LEVEL(1)  DO_LEVEL(2)  DO_LEVEL(3)  DO_LEVEL(4)
  DO_LEVEL(5)  DO_LEVEL(6)  DO_LEVEL(7)  DO_LEVEL(8)
  DO_LEVEL(9)  DO_LEVEL(10) DO_LEVEL(11) DO_LEVEL(12)
  DO_LEVEL(13) DO_LEVEL(14) DO_LEVEL(15)
#undef DO_LEVEL
  __syncthreads();

  // ---- Coalesced write-out: lane-contiguous float4 stores ---------------
  float* outBlk = out + (size_t)blockIdx.x * (BLOCK * 32);
#pragma unroll
  for (int k = 0; k < 8; ++k) {
    const int j = k * 1024 + tid * 4;  // 0..8191, float4 aligned
    const int p = j >> 5;              // point within block
    const int c = j & 31;              // output column
    float4 v;
    v.x = stage[p * 33 + c + 0];
    v.y = stage[p * 33 + c + 1];
    v.z = stage[p * 33 + c + 2];
    v.w = stage[p * 33 + c + 3];
    *reinterpret_cast<float4*>(outBlk + j) = v;
  }
}

extern "C" void kernel_launch(void* const* d_in, const int* in_sizes, int n_in,
                              void* d_out, int out_size, void* d_ws, size_t ws_size,
                              hipStream_t stream) {
  (void)in_sizes; (void)n_in; (void)out_size; (void)d_ws; (void)ws_size;
  const float* positions = (const float*)d_in[0];   // [524288, 3] f32
  const float* table     = (const float*)d_in[1];   // [11445040] f32
  float*       out       = (float*)d_out;           // [524288, 32] f32

  dim3 grid(NPOINTS / BLOCK);
  dim3 block(BLOCK);
  hipLaunchKernelGGL(hashenc_kernel, grid, block, 0, stream,
                     positions, table, out);
}